// BoxMpcController_29678224015735
// MI455X (gfx1250) — compile-verified
//
// CDNA5 / gfx1250 fused batched box-MPC IPM solver.
// One kernel launch runs all 12 IPM iterations. 32 workgroups x 128 threads
// (4 wave32s); each wave owns one batch element entirely in LDS (~60KB/WG).
// KKT solved via Schur complement -> block-tridiagonal (16 stages of 8x8)
// recursion; the 8x8 matrix products in the recursion use
// V_WMMA_F32_16X16X4_F32 (two WMMAs per 8x8x8 product, top-left quadrant).
#include <hip/hip_runtime.h>

typedef __attribute__((ext_vector_type(2))) float v2f;
typedef __attribute__((ext_vector_type(8))) float v8f;

namespace {

constexpr int XD = 8, UD = 4, SD = 2, HOR = 16;
constexpr int NBATCH = 128;
constexpr int NZ  = HOR * (XD + UD);  // 192
constexpr int NX  = HOR * XD;         // 128
constexpr int NU  = HOR * UD;         // 64
constexpr int MC  = 2 * NU;           // 128 inequality rows
constexpr int BPW = 4;                // batch elems per workgroup (1 per wave)
constexpr int NTHREADS = 32 * BPW;    // 128
constexpr int ITERS = 12;
constexpr float SIG = 0.1f;

struct PerBatch {
  float z[NZ], s[MC], lam[MC], nu[NX], b[NX];
  float rhs1[NZ], rp[NX], ri[MC], dv[MC];
  float g[NZ], y[NX], dnu[NX], dz[NZ], ds[MC], dlam[MC];
  float W[HOR * 16];     // per-stage inv(R + diag(e_t)), 4x4 row-major
  float Dinv[HOR * 64];  // per-stage 8x8 inverses for back-substitution
  float G1[64], T[64], aug[128], tmp8[8];
};

struct SMem {
  float A[64], B[32], C[16], Q[64], R[16];
  float Qi[64], AQi[64], QiAT[64], Tb1[64];  // Qi=Q^-1, Tb1=Qi+A Qi A^T
  float mu[BPW], alpha[BPW];
  PerBatch p[BPW];
};

// 8x8x8 f32 product P = X*Y packed into the top-left quadrant of a
// 16x16x4 f32 WMMA (two K-steps). Layouts per CDNA5 ISA 7.12.2:
//  A (16x4, 2 VGPR): lanes 0-15 rows, VGPR0/1 = K,K+1; lanes 16-31 = K+2,K+3.
//  B (4x16, 2 VGPR): mirrored (lanes 0-15 cols with K rows 0/1, 16-31 -> 2/3).
//  D (16x16, 8 VGPR): VGPR r = row r in lanes 0-15 (cols = lane).
// D rows 0-7 depend only on A rows 0-7 and B cols 0-7, and the two K-steps
// cover K=0..7 exactly, so A rows 8-15 / B cols 8-15 may hold ANY value: we
// mirror rows/cols via (lane & 7) so every LDS load is unconditional (no
// exec-masked single-element loads; adjacent A elements fuse to ds_load_b64).
__device__ inline v8f wmma_mul8(const float* __restrict__ X,
                                const float* __restrict__ Y, int lane) {
  v8f acc = {0.f, 0.f, 0.f, 0.f, 0.f, 0.f, 0.f, 0.f};
  const int r  = lane & 7;              // mirrored row/col index
  const int hi = (lane >> 4) ? 2 : 0;   // upper half-wave covers K+2,K+3
#pragma unroll
  for (int h = 0; h < 2; ++h) {
    const int k0 = 4 * h + hi;
    v2f af, bf;
    af.x = X[r * 8 + k0 + 0];    // A[r][k0]
    af.y = X[r * 8 + k0 + 1];    // A[r][k0+1]
    bf.x = Y[(k0 + 0) * 8 + r];  // B[k0][r]
    bf.y = Y[(k0 + 1) * 8 + r];  // B[k0+1][r]
    acc = __builtin_amdgcn_wmma_f32_16x16x4_f32(false, af, false, bf,
                                                (short)0, acc, false, false);
  }
  return acc;
}

// Gauss-Jordan inversion of an SPD 8x8 using 8 threads of each active wave.
// Called by ALL threads (block barriers inside); `act` is wave-uniform.
__device__ void gj_inv8(float* aug, float* out, const float* src, bool act,
                        int lr) {
  if (act) {
    for (int e = lr; e < 128; e += 32) {
      const int rr = e >> 4, cc = e & 15;
      aug[e] = (cc < 8) ? src[rr * 8 + cc] : ((cc - 8) == rr ? 1.f : 0.f);
    }
  }
  __syncthreads();
  for (int p = 0; p < 8; ++p) {
    if (act && lr < 16) {
      const float piv = aug[p * 16 + p];  // loads precede stores (same wave)
      aug[p * 16 + lr] /= piv;
    }
    __syncthreads();
    if (act && lr < 8 && lr != p) {
      const float f = aug[lr * 16 + p];
      for (int c = 0; c < 16; ++c) aug[lr * 16 + c] -= f * aug[p * 16 + c];
    }
    __syncthreads();
  }
  if (act) {
    for (int e = lr; e < 64; e += 32) out[e] = aug[(e >> 3) * 16 + 8 + (e & 7)];
  }
  __syncthreads();
}

}  // namespace

__global__ __launch_bounds__(NTHREADS) void mpc_ipm_kernel(
    const float* __restrict__ x0, const float* __restrict__ forecast,
    const float* __restrict__ Ag, const float* __restrict__ Bg,
    const float* __restrict__ Cg, const float* __restrict__ Qg,
    const float* __restrict__ Rg, float* __restrict__ out) {
  __shared__ SMem sm;
  const int tid = threadIdx.x;
  const int lane = tid & 31;
  const int wb = tid >> 5;  // wave index == local batch index

  // ---- constants into LDS ----
  for (int i = tid; i < 64; i += NTHREADS) sm.A[i] = Ag[i];
  for (int i = tid; i < 32; i += NTHREADS) sm.B[i] = Bg[i];
  for (int i = tid; i < 16; i += NTHREADS) sm.C[i] = Cg[i];
  for (int i = tid; i < 64; i += NTHREADS) sm.Q[i] = Qg[i];
  for (int i = tid; i < 16; i += NTHREADS) sm.R[i] = Rg[i];
  __syncthreads();

  gj_inv8(sm.p[0].aug, sm.Qi, sm.Q, wb == 0, lane);  // Qi = Q^-1

  for (int e = tid; e < 64; e += NTHREADS) {
    const int i = e >> 3, j = e & 7;
    float acc = 0.f;
    for (int k = 0; k < 8; ++k) acc += sm.A[i * 8 + k] * sm.Qi[k * 8 + j];
    sm.AQi[e] = acc;
  }
  __syncthreads();
  for (int e = tid; e < 64; e += NTHREADS) {
    const int i = e >> 3, j = e & 7;
    sm.QiAT[e] = sm.AQi[j * 8 + i];  // Qi symmetric -> Qi A^T = (A Qi)^T
    float acc = 0.f;
    for (int k = 0; k < 8; ++k) acc += sm.AQi[i * 8 + k] * sm.A[j * 8 + k];
    sm.Tb1[e] = sm.Qi[e] + acc;      // Qi + A Qi A^T
  }
  __syncthreads();

  PerBatch* P = &sm.p[wb];
  const int gb = blockIdx.x * BPW + wb;

  // ---- init state and b_t = C s_t (+ A x0 at t=0) ----
  for (int i = lane; i < NZ; i += 32) P->z[i] = 0.f;
  for (int i = lane; i < MC; i += 32) { P->s[i] = 1.f; P->lam[i] = 1.f; }
  for (int i = lane; i < NX; i += 32) P->nu[i] = 0.f;
  for (int i = lane; i < NX; i += 32) {
    const int t = i >> 3, r = i & 7;
    float acc = 0.f;
    for (int q = 0; q < SD; ++q)
      acc += sm.C[r * SD + q] * forecast[gb * HOR * SD + t * SD + q];
    if (t == 0)
      for (int j = 0; j < XD; ++j) acc += sm.A[r * 8 + j] * x0[gb * XD + j];
    P->b[i] = acc;
  }
  __syncthreads();

  for (int it = 0; it < ITERS; ++it) {
    // ---- mu ----
    if (lane == 0) {
      float acc = 0.f;
      for (int i = 0; i < MC; ++i) acc += P->s[i] * P->lam[i];
      sm.mu[wb] = acc * (1.f / MC);
    }
    __syncthreads();
    const float smu = SIG * sm.mu[wb];

    // ---- d = lam/s, r_i ----
    for (int i = lane; i < MC; i += 32) {
      P->dv[i] = P->lam[i] / P->s[i];
      const float u = (i < NU) ? P->z[NX + i] : -P->z[NX + i - NU];
      P->ri[i] = u + P->s[i] - 1.f;
    }
    // ---- rhs1 (x part) = -r_d_x, and r_p ----
    for (int i = lane; i < NX; i += 32) {
      const int t = i >> 3, r = i & 7;
      float rd = P->nu[i];
      for (int j = 0; j < 8; ++j) rd += sm.Q[r * 8 + j] * P->z[t * 8 + j];
      if (t < HOR - 1)
        for (int j = 0; j < 8; ++j) rd -= sm.A[j * 8 + r] * P->nu[(t + 1) * 8 + j];
      P->rhs1[i] = -rd;
      float rp = P->z[i] - P->b[i];
      if (t > 0)
        for (int j = 0; j < 8; ++j) rp -= sm.A[r * 8 + j] * P->z[(t - 1) * 8 + j];
      for (int l = 0; l < UD; ++l) rp -= sm.B[r * UD + l] * P->z[NX + t * UD + l];
      P->rp[i] = rp;
    }
    __syncthreads();
    // ---- rhs1 (u part) = -(r_d_u + w_j - w_{j+NU}) ----
    for (int i = lane; i < NU; i += 32) {
      const int t = i >> 2, r = i & 3;
      float rd = P->lam[i] - P->lam[NU + i];
      for (int l = 0; l < UD; ++l) rd += sm.R[r * UD + l] * P->z[NX + t * UD + l];
      for (int k = 0; k < XD; ++k) rd -= sm.B[k * UD + r] * P->nu[t * 8 + k];
      const float w0 = (P->lam[i] * P->ri[i] - (P->s[i] * P->lam[i] - smu)) / P->s[i];
      const float w1 = (P->lam[NU + i] * P->ri[NU + i] -
                        (P->s[NU + i] * P->lam[NU + i] - smu)) / P->s[NU + i];
      P->rhs1[NX + i] = -(rd + w0 - w1);
    }
    // ---- W_t = inv(R + diag(d_j + d_{j+NU})), one 4x4 per lane ----
    for (int t = lane; t < HOR; t += 32) {
      float a[4][8];
      for (int r = 0; r < 4; ++r)
        for (int c = 0; c < 4; ++c) {
          float v = sm.R[r * 4 + c];
          if (r == c) v += P->dv[t * UD + r] + P->dv[NU + t * UD + r];
          a[r][c] = v;
          a[r][4 + c] = (r == c) ? 1.f : 0.f;
        }
      for (int pv = 0; pv < 4; ++pv) {
        const float inv = 1.f / a[pv][pv];
        for (int c = 0; c < 8; ++c) a[pv][c] *= inv;
        for (int r = 0; r < 4; ++r)
          if (r != pv) {
            const float f = a[r][pv];
            for (int c = 0; c < 8; ++c) a[r][c] -= f * a[pv][c];
          }
      }
      for (int r = 0; r < 4; ++r)
        for (int c = 0; c < 4; ++c) P->W[t * 16 + r * 4 + c] = a[r][4 + c];
    }
    __syncthreads();

    // ---- g = M^-1 rhs1 ----
    for (int i = lane; i < NX; i += 32) {
      const int t = i >> 3, r = i & 7;
      float acc = 0.f;
      for (int j = 0; j < 8; ++j) acc += sm.Qi[r * 8 + j] * P->rhs1[t * 8 + j];
      P->g[i] = acc;
    }
    for (int i = lane; i < NU; i += 32) {
      const int t = i >> 2, r = i & 3;
      float acc = 0.f;
      for (int l = 0; l < UD; ++l)
        acc += P->W[t * 16 + r * 4 + l] * P->rhs1[NX + t * UD + l];
      P->g[NX + i] = acc;
    }
    __syncthreads();
    // ---- Schur rhs: y = Am g + r_p ----
    for (int i = lane; i < NX; i += 32) {
      const int t = i >> 3, r = i & 7;
      float acc = P->g[i] + P->rp[i];
      if (t > 0)
        for (int j = 0; j < 8; ++j) acc -= sm.A[r * 8 + j] * P->g[(t - 1) * 8 + j];
      for (int l = 0; l < UD; ++l) acc -= sm.B[r * UD + l] * P->g[NX + t * UD + l];
      P->y[i] = acc;
    }
    __syncthreads();

    // ---- block tridiagonal factor + forward sweep ----
    for (int t = 0; t < HOR; ++t) {
      // T_t = (t? Qi+AQiA^T : Qi) + B W_t B^T
      for (int e = lane; e < 64; e += 32) {
        const int i = e >> 3, j = e & 7;
        float acc = (t > 0) ? sm.Tb1[e] : sm.Qi[e];
        const float* Wt = &P->W[t * 16];
        for (int k = 0; k < UD; ++k) {
          float bw = 0.f;
          for (int l = 0; l < UD; ++l) bw += sm.B[i * UD + l] * Wt[l * 4 + k];
          acc += bw * sm.B[j * UD + k];
        }
        P->T[e] = acc;
      }
      __syncthreads();
      if (t > 0) {
        {  // G1 = AQi * Dinv[t-1]  (= -F_t), via f32 WMMA
          v8f acc = wmma_mul8(sm.AQi, &P->Dinv[(t - 1) * 64], lane);
          if (lane < 8) {
#pragma unroll
            for (int r = 0; r < 8; ++r) P->G1[r * 8 + lane] = acc[r];
          }
        }
        __syncthreads();
        {  // D_t = T_t - G1 * (Qi A^T), via f32 WMMA
          v8f acc = wmma_mul8(P->G1, sm.QiAT, lane);
          if (lane < 8) {
#pragma unroll
            for (int r = 0; r < 8; ++r) P->T[r * 8 + lane] -= acc[r];
          }
        }
        __syncthreads();
        if (lane < 8) {  // y_t += G1 * y_{t-1}
          float acc = 0.f;
          for (int j = 0; j < 8; ++j)
            acc += P->G1[lane * 8 + j] * P->y[(t - 1) * 8 + j];
          P->y[t * 8 + lane] += acc;
        }
        __syncthreads();
      }
      gj_inv8(P->aug, &P->Dinv[t * 64], P->T, true, lane);
    }

    // ---- backward sweep: dnu ----
    if (lane < 8) {
      float acc = 0.f;
      for (int j = 0; j < 8; ++j)
        acc += P->Dinv[(HOR - 1) * 64 + lane * 8 + j] * P->y[(HOR - 1) * 8 + j];
      P->dnu[(HOR - 1) * 8 + lane] = acc;
    }
    __syncthreads();
    for (int t = HOR - 2; t >= 0; --t) {
      if (lane < 8) {
        float acc = P->y[t * 8 + lane];
        for (int j = 0; j < 8; ++j)
          acc += sm.QiAT[lane * 8 + j] * P->dnu[(t + 1) * 8 + j];
        P->tmp8[lane] = acc;
      }
      __syncthreads();
      if (lane < 8) {
        float acc = 0.f;
        for (int j = 0; j < 8; ++j)
          acc += P->Dinv[t * 64 + lane * 8 + j] * P->tmp8[j];
        P->dnu[t * 8 + lane] = acc;
      }
      __syncthreads();
    }

    // ---- dz = M^-1 (rhs1 - Am^T dnu) ----
    for (int i = lane; i < NX; i += 32) {
      const int t = i >> 3, r = i & 7;
      float v = P->rhs1[i] - P->dnu[i];
      if (t < HOR - 1)
        for (int j = 0; j < 8; ++j) v += sm.A[j * 8 + r] * P->dnu[(t + 1) * 8 + j];
      P->g[i] = v;
    }
    for (int i = lane; i < NU; i += 32) {
      const int t = i >> 2, r = i & 3;
      float v = P->rhs1[NX + i];
      for (int k = 0; k < XD; ++k) v += sm.B[k * UD + r] * P->dnu[t * 8 + k];
      P->g[NX + i] = v;
    }
    __syncthreads();
    for (int i = lane; i < NX; i += 32) {
      const int t = i >> 3, r = i & 7;
      float acc = 0.f;
      for (int j = 0; j < 8; ++j) acc += sm.Qi[r * 8 + j] * P->g[t * 8 + j];
      P->dz[i] = acc;
    }
    for (int i = lane; i < NU; i += 32) {
      const int t = i >> 2, r = i & 3;
      float acc = 0.f;
      for (int l = 0; l < UD; ++l)
        acc += P->W[t * 16 + r * 4 + l] * P->g[NX + t * UD + l];
      P->dz[NX + i] = acc;
    }
    __syncthreads();

    // ---- ds, dlam ----
    for (int i = lane; i < MC; i += 32) {
      const float gdz = (i < NU) ? P->dz[NX + i] : -P->dz[NX + i - NU];
      const float dsv = -P->ri[i] - gdz;
      const float rc = P->s[i] * P->lam[i] - smu;
      P->ds[i] = dsv;
      P->dlam[i] = (-rc - P->lam[i] * dsv) / P->s[i];
    }
    __syncthreads();
    // ---- fraction-to-boundary step ----
    if (lane == 0) {
      float rs = 3.0e38f, rl = 3.0e38f;
      for (int i = 0; i < MC; ++i) {
        if (P->ds[i] < 0.f) rs = fminf(rs, -P->s[i] / P->ds[i]);
        if (P->dlam[i] < 0.f) rl = fminf(rl, -P->lam[i] / P->dlam[i]);
      }
      sm.alpha[wb] = fminf(fminf(1.f, 0.99f * rs), fminf(1.f, 0.99f * rl));
    }
    __syncthreads();
    const float a = sm.alpha[wb];
    for (int i = lane; i < NZ; i += 32) P->z[i] += a * P->dz[i];
    for (int i = lane; i < MC; i += 32) {
      P->s[i] += a * P->ds[i];
      P->lam[i] += a * P->dlam[i];
    }
    for (int i = lane; i < NX; i += 32) P->nu[i] += a * P->dnu[i];
    __syncthreads();
  }

  // ---- outputs: x_sol (BATCH,16,8) then u_sol (BATCH,16,4), flat ----
  for (int i = lane; i < NX; i += 32) out[gb * NX + i] = P->z[i];
  for (int i = lane; i < NU; i += 32)
    out[NBATCH * NX + gb * NU + i] = P->z[NX + i];
}

extern "C" void kernel_launch(void* const* d_in, const int* in_sizes, int n_in,
                              void* d_out, int out_size, void* d_ws,
                              size_t ws_size, hipStream_t stream) {
  (void)in_sizes; (void)n_in; (void)out_size; (void)d_ws; (void)ws_size;
  const float* x0 = (const float*)d_in[0];
  const float* fc = (const float*)d_in[1];
  const float* A  = (const float*)d_in[2];
  const float* B  = (const float*)d_in[3];
  const float* C  = (const float*)d_in[4];
  const float* Q  = (const float*)d_in[5];
  const float* R  = (const float*)d_in[6];
  float* out = (float*)d_out;
  dim3 grid(NBATCH / BPW), block(NTHREADS);
  hipLaunchKernelGGL(mpc_ipm_kernel, grid, block, 0, stream, x0, fc, A, B, C,
                     Q, R, out);
}